// AttentionLayer_45792941310015
// MI455X (gfx1250) — compile-verified
//
#include <hip/hip_runtime.h>

typedef __attribute__((ext_vector_type(2))) float v2f;
typedef __attribute__((ext_vector_type(8))) float v8f;

#define Bn 16
#define Sn 4096
#define Vn 1024

// ---------------------------------------------------------------------------
// Zero the accumulator regions of the workspace (w: Bn*Vn floats, A: Bn floats)
// ---------------------------------------------------------------------------
__global__ __launch_bounds__(256) void zero_ws_kernel(float* __restrict__ w,
                                                      float* __restrict__ Asum) {
    int i = blockIdx.x * 256 + threadIdx.x;
    if (i < Bn * Vn) w[i] = 0.0f;
    if (i < Bn) Asum[i] = 0.0f;
}

// ---------------------------------------------------------------------------
// q[b,n] = sum_k pool[b,k] * Wq[n,k] + bq[n]
// One wave per block; block x = 16-wide N tile. M=16 covers all batches.
// Uses V_WMMA_F32_16X16X4_F32: A-frag v2f, B-frag v2f, C/D v8f.
//   A (16x4): lanes 0-15 -> M=lane, K=k0+{0,1}; lanes 16-31 -> K=k0+{2,3}
//   B (4x16): lanes 0-15 -> N=lane, K=k0+{0,1}; lanes 16-31 -> K=k0+{2,3}
//   D: VGPR r -> M = r + (lane>=16 ? 8:0), N = lane&15
// ---------------------------------------------------------------------------
__global__ __launch_bounds__(32) void q_gemm_kernel(const float* __restrict__ pool,
                                                    const float* __restrict__ Wq,
                                                    const float* __restrict__ bq,
                                                    float* __restrict__ q) {
    const int n0   = blockIdx.x * 16;
    const int lane = threadIdx.x;
    const int m    = lane & 15;
    const int kk   = (lane >> 4) << 1;  // 0 or 2
    const int n    = n0 + m;

    v8f acc = {};
#pragma unroll 4
    for (int k0 = 0; k0 < Vn; k0 += 4) {
        v2f a = *(const v2f*)(pool + m * Vn + k0 + kk);
        v2f b = *(const v2f*)(Wq + n * Vn + k0 + kk);  // Bmat[k][n] = Wq[n][k]
        acc = __builtin_amdgcn_wmma_f32_16x16x4_f32(false, a, false, b,
                                                    (short)0, acc, false, false);
    }
    const float bias  = bq[n];
    const int   mbase = (lane >> 4) << 3;  // 0 or 8
#pragma unroll
    for (int r = 0; r < 8; ++r) q[(mbase + r) * Vn + n] = acc[r] + bias;
}

// ---------------------------------------------------------------------------
// u[b,n] = sum_k q[b,k] * Wk[k,n]   (note: NOT transposed access)
// c[b]   = sum_o q[b,o] * bk[o]
// ---------------------------------------------------------------------------
__global__ __launch_bounds__(32) void u_gemm_kernel(const float* __restrict__ q,
                                                    const float* __restrict__ Wk,
                                                    const float* __restrict__ bk,
                                                    float* __restrict__ u,
                                                    float* __restrict__ cvec) {
    const int n0   = blockIdx.x * 16;
    const int lane = threadIdx.x;
    const int m    = lane & 15;
    const int kk   = (lane >> 4) << 1;
    const int n    = n0 + m;

    v8f acc = {};
#pragma unroll 4
    for (int k0 = 0; k0 < Vn; k0 += 4) {
        v2f a = *(const v2f*)(q + m * Vn + k0 + kk);
        v2f b;
        b.x = Wk[(k0 + kk) * Vn + n];      // Bmat[k][n] = Wk[k][n]
        b.y = Wk[(k0 + kk + 1) * Vn + n];
        acc = __builtin_amdgcn_wmma_f32_16x16x4_f32(false, a, false, b,
                                                    (short)0, acc, false, false);
    }
    const int mbase = (lane >> 4) << 3;
#pragma unroll
    for (int r = 0; r < 8; ++r) u[(mbase + r) * Vn + n] = acc[r];

    if (blockIdx.x == 0 && lane < 16) {  // tiny: c[b] = q[b,:] . bk
        float s = 0.0f;
        for (int o = 0; o < Vn; ++o) s += q[lane * Vn + o] * bk[o];
        cvec[lane] = s;
    }
}

// ---------------------------------------------------------------------------
// Single streaming pass over bert (the 256 MiB, bandwidth-bound part):
//   a_s = (u . x_s + c) / 64 ;  w += a_s * x_s ;  A += a_s
// grid = (16 blocks, Bn batches), 256 threads = 8 waves.
// Each wave owns 32 contiguous rows; lanes cover the full 1024 columns
// (8 chunks of float4, chunk stride 128 for coalescing).
// ---------------------------------------------------------------------------
__global__ __launch_bounds__(256) void attn_stream_kernel(const float* __restrict__ bert,
                                                          const float* __restrict__ u,
                                                          const float* __restrict__ cvec,
                                                          float* __restrict__ w,
                                                          float* __restrict__ Asum) {
    __shared__ float lw[Vn];
    const int b    = blockIdx.y;
    const int tid  = threadIdx.x;
    const int lane = tid & 31;
    const int wv   = blockIdx.x * 8 + (tid >> 5);  // 0..127 wave id within batch

    for (int i = tid; i < Vn; i += 256) lw[i] = 0.0f;
    __syncthreads();

    float4 uf[8];
#pragma unroll
    for (int ch = 0; ch < 8; ++ch)
        uf[ch] = *(const float4*)(u + b * Vn + ch * 128 + lane * 4);
    const float cb = cvec[b];

    float4 wacc[8];
#pragma unroll
    for (int ch = 0; ch < 8; ++ch) wacc[ch] = make_float4(0.f, 0.f, 0.f, 0.f);
    float asum = 0.0f;

    const float* bbase = bert + (size_t)b * Sn * Vn;
    const int    s0    = wv * 32;
    for (int si = 0; si < 32; ++si) {
        const float* row = bbase + (size_t)(s0 + si) * Vn;
        __builtin_prefetch(row + 2 * Vn + lane * 4, 0, 0);  // global_prefetch_b8
        float4 x[8];
        float  p = 0.0f;
#pragma unroll
        for (int ch = 0; ch < 8; ++ch) {
            x[ch] = *(const float4*)(row + ch * 128 + lane * 4);
            p += x[ch].x * uf[ch].x + x[ch].y * uf[ch].y +
                 x[ch].z * uf[ch].z + x[ch].w * uf[ch].w;
        }
#pragma unroll
        for (int off = 16; off > 0; off >>= 1) p += __shfl_xor(p, off, 32);
        const float a = (p + cb) * 0.015625f;  // 1/sqrt(4096)
        asum += a;
#pragma unroll
        for (int ch = 0; ch < 8; ++ch) {
            wacc[ch].x += a * x[ch].x;
            wacc[ch].y += a * x[ch].y;
            wacc[ch].z += a * x[ch].z;
            wacc[ch].w += a * x[ch].w;
        }
    }

    // combine the 8 waves via LDS float atomics (ds_add_f32)
#pragma unroll
    for (int ch = 0; ch < 8; ++ch) {
        const int j = ch * 128 + lane * 4;
        atomicAdd(&lw[j + 0], wacc[ch].x);
        atomicAdd(&lw[j + 1], wacc[ch].y);
        atomicAdd(&lw[j + 2], wacc[ch].z);
        atomicAdd(&lw[j + 3], wacc[ch].w);
    }
    if (lane == 0) atomicAdd(&Asum[b], asum);
    __syncthreads();
    for (int i = tid; i < Vn; i += 256) atomicAdd(&w[b * Vn + i], lw[i]);
}

// ---------------------------------------------------------------------------
// out[b,n] = sum_k w[b,k] * Wv[n,k] + A[b] * bv[n]
// ---------------------------------------------------------------------------
__global__ __launch_bounds__(32) void out_gemm_kernel(const float* __restrict__ w,
                                                      const float* __restrict__ Wv,
                                                      const float* __restrict__ bv,
                                                      const float* __restrict__ Asum,
                                                      float* __restrict__ out) {
    const int n0   = blockIdx.x * 16;
    const int lane = threadIdx.x;
    const int m    = lane & 15;
    const int kk   = (lane >> 4) << 1;
    const int n    = n0 + m;

    v8f acc = {};
#pragma unroll 4
    for (int k0 = 0; k0 < Vn; k0 += 4) {
        v2f a = *(const v2f*)(w + m * Vn + k0 + kk);
        v2f b = *(const v2f*)(Wv + n * Vn + k0 + kk);  // Bmat[k][n] = Wv[n][k]
        acc = __builtin_amdgcn_wmma_f32_16x16x4_f32(false, a, false, b,
                                                    (short)0, acc, false, false);
    }
    const float bvn   = bv[n];
    const int   mbase = (lane >> 4) << 3;
#pragma unroll
    for (int r = 0; r < 8; ++r) {
        const int mm = mbase + r;
        out[mm * Vn + n] = acc[r] + Asum[mm] * bvn;
    }
}

// ---------------------------------------------------------------------------
extern "C" void kernel_launch(void* const* d_in, const int* in_sizes, int n_in,
                              void* d_out, int out_size, void* d_ws, size_t ws_size,
                              hipStream_t stream) {
    const float* pool = (const float*)d_in[0];  // [16,1,1024]
    const float* bert = (const float*)d_in[1];  // [16,4096,1024]
    const float* Wq   = (const float*)d_in[2];
    const float* bq   = (const float*)d_in[3];
    const float* Wk   = (const float*)d_in[4];
    const float* bk   = (const float*)d_in[5];
    const float* Wv   = (const float*)d_in[6];
    const float* bv   = (const float*)d_in[7];
    float*       out  = (float*)d_out;

    float* ws = (float*)d_ws;
    float* q  = ws;                 // Bn*Vn
    float* u  = ws + Bn * Vn;       // Bn*Vn
    float* w  = ws + 2 * Bn * Vn;   // Bn*Vn
    float* cv = ws + 3 * Bn * Vn;   // Bn
    float* As = ws + 3 * Bn * Vn + Bn;

    zero_ws_kernel<<<(Bn * Vn) / 256, 256, 0, stream>>>(w, As);
    q_gemm_kernel<<<Vn / 16, 32, 0, stream>>>(pool, Wq, bq, q);
    u_gemm_kernel<<<Vn / 16, 32, 0, stream>>>(q, Wk, bk, u, cv);
    attn_stream_kernel<<<dim3(16, Bn), 256, 0, stream>>>(bert, u, cv, w, As);
    out_gemm_kernel<<<Vn / 16, 32, 0, stream>>>(w, Wv, bv, As, out);
}